// BeliefPropagation_16303695855738
// MI455X (gfx1250) — compile-verified
//
#include <hip/hip_runtime.h>
#include <math.h>

#define Q 16
#define BP_MAX_DIFF 0.01f
#define MAX_ITER 10

typedef float v2f __attribute__((ext_vector_type(2)));
typedef float v8f __attribute__((ext_vector_type(8)));

#define WMMA_F32X4(A, B, C) \
    __builtin_amdgcn_wmma_f32_16x16x4_f32(false, (A), false, (B), (short)0, (C), false, false)

// ---------------------------------------------------------------------------
// consts[0] = exp(beta)-1 ; consts[1] = beta
__global__ void k_init_consts(const float* __restrict__ beta, float* __restrict__ consts) {
    if (blockIdx.x == 0 && threadIdx.x == 0) {
        float b = beta[0];
        consts[0] = expf(b) - 1.0f;
        consts[1] = b;
    }
}

// ---------------------------------------------------------------------------
// Weighted column sum  out[q] += sum_n w[n]*psi[n,q]  via V_WMMA_F32_16X16X4_F32.
// A = ones(16x4), B = 4 rows of psi (weighted). All-ones A makes the result
// invariant to the K-slot permutation of the B VGPR layout.
// Main loop is guard-free (full 4-row chunks, coalesced b32 loads); tail rows
// are handled by wave 0 with guarded loads. Two accumulator chains so the XDL
// pipe can overlap back-to-back WMMAs.
__global__ void k_colsum_wmma(const float* __restrict__ psi,
                              const float* __restrict__ w,   // may be nullptr
                              float* __restrict__ out, int n_rows) {
    const int lane   = threadIdx.x & 31;
    const int wave   = blockIdx.x * (blockDim.x >> 5) + (threadIdx.x >> 5);
    const int nWaves = gridDim.x * (blockDim.x >> 5);
    const int nFull  = n_rows >> 2;          // full 4-row chunks

    v8f acc0 = {}, acc1 = {};
    v2f a;  a.x = 1.0f; a.y = 1.0f;          // ones A-matrix

    int c = wave;                            // uniform per wave
    if (w == nullptr) {
        for (; c + nWaves < nFull; c += 2 * nWaves) {
            const size_t o0 = (size_t)c << 6;            // chunk c * 64 floats
            const size_t o1 = (size_t)(c + nWaves) << 6;
            v2f b0, b1;
            b0.x = psi[o0 + lane];  b0.y = psi[o0 + 32 + lane];
            b1.x = psi[o1 + lane];  b1.y = psi[o1 + 32 + lane];
            acc0 = WMMA_F32X4(a, b0, acc0);
            acc1 = WMMA_F32X4(a, b1, acc1);
        }
        if (c < nFull) {
            const size_t o = (size_t)c << 6;
            v2f b;  b.x = psi[o + lane];  b.y = psi[o + 32 + lane];
            acc0 = WMMA_F32X4(a, b, acc0);
        }
    } else {
        for (; c + nWaves < nFull; c += 2 * nWaves) {
            const size_t o0 = (size_t)c << 6;
            const size_t o1 = (size_t)(c + nWaves) << 6;
            const int r0 = (c << 2) + (lane >> 4);
            const int r1 = ((c + nWaves) << 2) + (lane >> 4);
            v2f b0, b1;
            b0.x = psi[o0 + lane]      * w[r0];
            b0.y = psi[o0 + 32 + lane] * w[r0 + 2];
            b1.x = psi[o1 + lane]      * w[r1];
            b1.y = psi[o1 + 32 + lane] * w[r1 + 2];
            acc0 = WMMA_F32X4(a, b0, acc0);
            acc1 = WMMA_F32X4(a, b1, acc1);
        }
        if (c < nFull) {
            const size_t o = (size_t)c << 6;
            const int r = (c << 2) + (lane >> 4);
            v2f b;
            b.x = psi[o + lane]      * w[r];
            b.y = psi[o + 32 + lane] * w[r + 2];
            acc0 = WMMA_F32X4(a, b, acc0);
        }
    }

    // tail rows (n_rows % 4 != 0): wave 0 only, guarded loads (uniform branch)
    if (wave == 0 && (n_rows & 3)) {
        const int rowbase = nFull << 2;
        const int r0 = rowbase + (lane >> 4);
        const int r1 = r0 + 2;
        float b0 = (r0 < n_rows) ? psi[(size_t)rowbase * Q + lane]      : 0.0f;
        float b1 = (r1 < n_rows) ? psi[(size_t)rowbase * Q + 32 + lane] : 0.0f;
        if (w) {
            b0 *= (r0 < n_rows) ? w[r0] : 0.0f;
            b1 *= (r1 < n_rows) ? w[r1] : 0.0f;
        }
        v2f b;  b.x = b0;  b.y = b1;
        acc0 = WMMA_F32X4(a, b, acc0);
    }

    // D row M=0 lives in vgpr0 of the accumulator, lanes 0..15 (N = lane)
    if (lane < Q) atomicAdd(&out[lane], acc0[0] + acc1[0]);
}

// ---------------------------------------------------------------------------
// Pass 1: node_log[dst] += log1p(msg * (e^b - 1))
__global__ void k_edge_pass1(const float* __restrict__ msg,
                             const int* __restrict__ dst,
                             float* __restrict__ nlog,
                             const float* __restrict__ consts,
                             const int* __restrict__ done, int n_edges) {
    if (*done) return;
    int e = blockIdx.x * blockDim.x + threadIdx.x;
    if (e >= n_edges) return;
    const float eb = consts[0];
    const int d = dst[e];
    const float4* mrow = (const float4*)(msg + (size_t)e * Q);
    float* nrow = nlog + (size_t)d * Q;
#pragma unroll
    for (int v = 0; v < 4; ++v) {
        float4 m = mrow[v];
        atomicAdd(&nrow[v * 4 + 0], log1pf(m.x * eb));
        atomicAdd(&nrow[v * 4 + 1], log1pf(m.y * eb));
        atomicAdd(&nrow[v * 4 + 2], log1pf(m.z * eb));
        atomicAdd(&nrow[v * 4 + 3], log1pf(m.w * eb));
    }
}

// ---------------------------------------------------------------------------
// Pass 2: cavity softmax -> msgB; diff max; node_log2[dst] += log1p(new*eb)
__global__ void k_edge_pass2(const float* __restrict__ msgA,
                             float* __restrict__ msgB,
                             const int* __restrict__ src,
                             const int* __restrict__ dst,
                             const int* __restrict__ rev,
                             const float* __restrict__ nlog,
                             float* __restrict__ nlog2,
                             const float* __restrict__ h,
                             const float* __restrict__ consts,
                             const int* __restrict__ done,
                             unsigned* __restrict__ diffbits, int n_edges) {
    int e = blockIdx.x * blockDim.x + threadIdx.x;
    if (e >= n_edges) return;
    const size_t eQ = (size_t)e * Q;
    if (*done) {   // frozen: keep ping-pong buffers consistent
        const float4* i = (const float4*)(msgA + eQ);
        float4* o = (float4*)(msgB + eQ);
        o[0] = i[0]; o[1] = i[1]; o[2] = i[2]; o[3] = i[3];
        return;
    }
    const float eb = consts[0];
    const int s = src[e];
    const int r = rev[e];
    const int d = dst[e];
    const float* nrow = nlog + (size_t)s * Q;
    const float* mrev = msgA + (size_t)r * Q;

    float old[Q];
    {
        const float4* mold = (const float4*)(msgA + eQ);
#pragma unroll
        for (int v = 0; v < 4; ++v) ((float4*)old)[v] = mold[v];
    }

    float t[Q];
    float mx = -1e30f;
#pragma unroll
    for (int q = 0; q < Q; ++q) {
        t[q] = nrow[q] - log1pf(mrev[q] * eb) + h[q];
        mx = fmaxf(mx, t[q]);
    }
    float sum = 0.0f;
#pragma unroll
    for (int q = 0; q < Q; ++q) { t[q] = expf(t[q] - mx); sum += t[q]; }
    const float inv = 1.0f / sum;

    float* n2 = nlog2 + (size_t)d * Q;
    float nm[Q];
    float diff = 0.0f;
#pragma unroll
    for (int q = 0; q < Q; ++q) {
        nm[q] = t[q] * inv;
        diff += fabsf(nm[q] - old[q]);
        atomicAdd(&n2[q], log1pf(nm[q] * eb));
    }
    {
        float4* o = (float4*)(msgB + eQ);
#pragma unroll
        for (int v = 0; v < 4; ++v) o[v] = ((float4*)nm)[v];
    }
    atomicMax(diffbits, __float_as_uint(diff));   // diff >= 0 -> bit-order OK
}

// ---------------------------------------------------------------------------
// Node pass: psi = softmax(node_log2 + h)
__global__ void k_node_pass(const float* __restrict__ nlog2,
                            const float* __restrict__ h,
                            float* __restrict__ psi,
                            const int* __restrict__ done, int n_nodes) {
    if (*done) return;
    int n = blockIdx.x * blockDim.x + threadIdx.x;
    if (n >= n_nodes) return;
    const size_t nQ = (size_t)n * Q;
    float t[Q];
    float mx = -1e30f;
#pragma unroll
    for (int q = 0; q < Q; ++q) { t[q] = nlog2[nQ + q] + h[q]; mx = fmaxf(mx, t[q]); }
    float sum = 0.0f;
#pragma unroll
    for (int q = 0; q < Q; ++q) { t[q] = expf(t[q] - mx); sum += t[q]; }
    const float inv = 1.0f / sum;
#pragma unroll
    for (int q = 0; q < Q; ++q) psi[nQ + q] = t[q] * inv;
}

// ---------------------------------------------------------------------------
// Per-iteration epilogue: h update (gated on old done) + done |= diff < eps
__global__ void k_iter_finalize(const float* __restrict__ colsum,
                                float* __restrict__ h,
                                const unsigned* __restrict__ diffbits,
                                int* __restrict__ done,
                                const float* __restrict__ consts,
                                float mean_w, int force_h) {
    const int q = threadIdx.x;
    const int was_done = *done;
    if (q < Q && (!was_done || force_h))
        h[q] = -consts[1] * mean_w * colsum[q];
    __syncthreads();
    if (q == 0 && !force_h) {
        if (__uint_as_float(*diffbits) < BP_MAX_DIFF) *done = 1;
    }
}

// ---------------------------------------------------------------------------
__global__ void k_deg(const int* __restrict__ dst, float* __restrict__ deg, int n_edges) {
    int e = blockIdx.x * blockDim.x + threadIdx.x;
    if (e < n_edges) atomicAdd(&deg[dst[e]], 1.0f);
}

__global__ void k_edge_mod(const float* __restrict__ psi,
                           const int* __restrict__ src,
                           const int* __restrict__ dst,
                           float* __restrict__ out_sum, int n_edges) {
    __shared__ float red[256];
    int e = blockIdx.x * blockDim.x + threadIdx.x;
    float s = 0.0f;
    if (e < n_edges) {
        const float* a = psi + (size_t)src[e] * Q;
        const float* b = psi + (size_t)dst[e] * Q;
#pragma unroll
        for (int q = 0; q < Q; ++q) s += a[q] * b[q];
    }
    red[threadIdx.x] = s;
    __syncthreads();
    for (int w = 128; w > 0; w >>= 1) {
        if (threadIdx.x < w) red[threadIdx.x] += red[threadIdx.x + w];
        __syncthreads();
    }
    if (threadIdx.x == 0) atomicAdd(out_sum, red[0]);
}

__global__ void k_node_ent(const float* __restrict__ psi,
                           float* __restrict__ out_sum, int n_nodes) {
    __shared__ float red[256];
    int n = blockIdx.x * blockDim.x + threadIdx.x;
    float s = 0.0f;
    if (n < n_nodes) {
        const float* p = psi + (size_t)n * Q;
#pragma unroll
        for (int q = 0; q < Q; ++q) s -= p[q] * logf(p[q] + 1e-12f);
    }
    red[threadIdx.x] = s;
    __syncthreads();
    for (int w = 128; w > 0; w >>= 1) {
        if (threadIdx.x < w) red[threadIdx.x] += red[threadIdx.x + w];
        __syncthreads();
    }
    if (threadIdx.x == 0) atomicAdd(out_sum, red[0]);
}

__global__ void k_final_scalars(const float* __restrict__ colsum,
                                const float* __restrict__ degcol,
                                const float* __restrict__ scal,
                                float* __restrict__ out3,
                                float n_nodesf, float n_edgesf) {
    if (blockIdx.x != 0 || threadIdx.x != 0) return;
    float reg = 0.0f;
#pragma unroll
    for (int q = 0; q < Q; ++q) {
        float d = colsum[q] / n_nodesf - 1.0f / (float)Q;
        reg += d * d;
    }
    reg *= sqrtf((float)Q);
    float ent = (scal[1] / n_nodesf) / logf((float)Q);
    float dt = 0.0f;
#pragma unroll
    for (int q = 0; q < Q; ++q) {
        float x = degcol[q] / n_edgesf;
        dt += x * x;
    }
    float mod = scal[0] / n_edgesf - dt;
    out3[0] = reg;   // group-balance regularizer
    out3[1] = ent;   // entropy loss
    out3[2] = mod;   // modularity
}

// ---------------------------------------------------------------------------
extern "C" void kernel_launch(void* const* d_in, const int* in_sizes, int n_in,
                              void* d_out, int out_size, void* d_ws, size_t ws_size,
                              hipStream_t stream) {
    const int*   src  = (const int*)d_in[0];
    const int*   dst  = (const int*)d_in[1];
    const int*   rev  = (const int*)d_in[2];
    const float* psi0 = (const float*)d_in[3];
    const float* msg0 = (const float*)d_in[4];
    const float* beta = (const float*)d_in[5];

    const int E = in_sizes[0];
    const int N = in_sizes[3] / Q;
    const float mean_w = (float)E / ((float)N * (float)N);

    const size_t msgElems  = (size_t)E * Q;
    const size_t nodeElems = (size_t)N * Q;

    float* ws       = (float*)d_ws;
    float* msgA     = ws;
    float* msgB     = msgA + msgElems;
    float* nlog     = msgB + msgElems;
    float* nlog2    = nlog + nodeElems;
    float* psi      = nlog2 + nodeElems;
    float* deg      = psi + nodeElems;          // N floats
    float* hbuf     = deg + N;                  // Q
    float* colsum   = hbuf + Q;                 // Q
    float* degcol   = colsum + Q;               // Q
    float* consts   = degcol + Q;               // 2
    unsigned* diffb = (unsigned*)(consts + 2);  // 1
    int*   done     = (int*)(diffb + 1);        // 1
    float* scal     = (float*)(done + 1);       // [0]=edge_sum [1]=ent_sum

    const int TB = 256;
    const int egrid = (E + TB - 1) / TB;
    const int ngrid = (N + TB - 1) / TB;

    // --- init state ---
    hipMemcpyAsync(msgA, msg0, msgElems * sizeof(float), hipMemcpyDeviceToDevice, stream);
    hipMemcpyAsync(psi,  psi0, nodeElems * sizeof(float), hipMemcpyDeviceToDevice, stream);
    hipMemsetAsync(deg, 0, (size_t)N * sizeof(float), stream);
    hipMemsetAsync(hbuf, 0, (size_t)((char*)(scal + 2) - (char*)hbuf), stream);

    k_init_consts<<<1, 1, 0, stream>>>(beta, consts);
    k_colsum_wmma<<<64, 256, 0, stream>>>(psi, nullptr, colsum, N);
    k_iter_finalize<<<1, 32, 0, stream>>>(colsum, hbuf, diffb, done, consts, mean_w, 1);

    // --- BP iterations (fixed launch count; device-side `done` gating) ---
    for (int it = 0; it < MAX_ITER; ++it) {
        hipMemsetAsync(nlog,  0, nodeElems * sizeof(float), stream);
        hipMemsetAsync(nlog2, 0, nodeElems * sizeof(float), stream);
        hipMemsetAsync(colsum, 0, Q * sizeof(float), stream);
        hipMemsetAsync(diffb, 0, sizeof(unsigned), stream);

        k_edge_pass1<<<egrid, TB, 0, stream>>>(msgA, dst, nlog, consts, done, E);
        k_edge_pass2<<<egrid, TB, 0, stream>>>(msgA, msgB, src, dst, rev,
                                               nlog, nlog2, hbuf, consts, done, diffb, E);
        k_node_pass<<<ngrid, TB, 0, stream>>>(nlog2, hbuf, psi, done, N);
        k_colsum_wmma<<<64, 256, 0, stream>>>(psi, nullptr, colsum, N);
        k_iter_finalize<<<1, 32, 0, stream>>>(colsum, hbuf, diffb, done, consts, mean_w, 0);

        float* t = msgA; msgA = msgB; msgB = t;
    }

    // --- final scalars ---
    k_deg<<<egrid, TB, 0, stream>>>(dst, deg, E);
    k_edge_mod<<<egrid, TB, 0, stream>>>(psi, src, dst, scal + 0, E);
    k_node_ent<<<ngrid, TB, 0, stream>>>(psi, scal + 1, N);
    k_colsum_wmma<<<64, 256, 0, stream>>>(psi, deg, degcol, N);

    float* out = (float*)d_out;
    hipMemcpyAsync(out, psi, nodeElems * sizeof(float), hipMemcpyDeviceToDevice, stream);
    k_final_scalars<<<1, 1, 0, stream>>>(colsum, degcol, scal, out + nodeElems,
                                         (float)N, (float)E);
}